// UFGConv_57913339019819
// MI455X (gfx1250) — compile-verified
//
#include <hip/hip_runtime.h>

#define F 128

typedef __attribute__((ext_vector_type(2))) float v2f;
typedef __attribute__((ext_vector_type(8))) float v8f;
typedef __attribute__((ext_vector_type(4))) unsigned int v4u;
typedef __attribute__((ext_vector_type(8))) int v8i;
typedef __attribute__((ext_vector_type(4))) int v4i;

// LDS row stride for staged W: 128 + 8 pad DWORDs => 2 rows apart = 272 DW,
// 272 % 64 = 16 banks shift between the two half-wave lane groups.
#define BSTRIDE 136

// ---------------------------------------------------------------------------
// Zero-fill (float4 vectorized)
// ---------------------------------------------------------------------------
__global__ void zero_kernel(float* __restrict__ p, long n) {
  long i = ((long)blockIdx.x * blockDim.x + threadIdx.x) * 4;
  long stride = (long)gridDim.x * blockDim.x * 4;
  for (long j = i; j < n; j += stride) {
    *(float4*)(p + j) = make_float4(0.f, 0.f, 0.f, 0.f);
  }
}

// ---------------------------------------------------------------------------
// Dense GEMM  C[M,128] = A[M,128] * B[128,128]  via V_WMMA_F32_16X16X4_F32.
// One wave per 16x16 tile; block = 8 waves -> one 16x128 output strip.
// W is DMA'd once per block into LDS by the Tensor Data Mover (single
// tensor_load_to_lds, TENSORcnt-tracked), with 8-DWORD/row LDS padding to
// kill bank conflicts on the B-fragment ds_loads.
// Fragment layouts per CDNA5 ISA 7.12.2:
//   A 16x4 : lane%16 = row M, vgpr i holds K = i + 2*(lane>=16)
//   B 4x16 : lane%16 = col N, vgpr i holds K = i + 2*(lane>=16)
//   C 16x16: lane%16 = col N, vgpr r holds row M = r + 8*(lane>=16)
// ---------------------------------------------------------------------------
__global__ void gemm_wmma_f32(const float* __restrict__ A,
                              const float* __restrict__ B,
                              float* __restrict__ C, int M) {
  __shared__ float sB[F * BSTRIDE];   // 128 rows x 136 floats = 69632 B

  const int lane = threadIdx.x & 31;
  const int wave = threadIdx.x >> 5;
  const int l    = lane & 15;
  const int hf   = lane >> 4;
  const int m0   = blockIdx.x * 16;
  const int n0   = wave * 16;

  // ---- TDM: stage W[128,128] f32 into LDS (wave 0 issues, all waves sync) --
  if (wave == 0) {
    unsigned long long gaddr = (unsigned long long)B;
    v4u g0;
    g0[0] = 1u;                                   // count=1, user descriptor
    g0[1] = 0u;                                   // lds_addr = 0 (sB base)
    g0[2] = (unsigned int)gaddr;                  // global_addr[31:0]
    g0[3] = ((unsigned int)(gaddr >> 32) & 0x01FFFFFFu) | 0x80000000u; // type=2
    v8i g1;
    // data_size=4B (2<<16) | pad_enable (1<<20) | pad_interval=6:128DW (6<<22)
    // | pad_amount=7:8DW (7<<25)
    g1[0] = (int)((2u << 16) | (1u << 20) | (6u << 22) | (7u << 25));
    g1[1] = (int)(128u << 16);                    // tensor_dim0 = 128
    g1[2] = (int)(128u << 16);                    // tensor_dim1 = 128
    g1[3] = (int)(128u << 16);                    // tile_dim0 = 128
    g1[4] = (int)128u;                            // tile_dim1 = 128
    g1[5] = (int)128u;                            // tensor_dim0_stride = 128
    g1[6] = 0; g1[7] = 0;
    v4i gz4 = {0, 0, 0, 0};
    v8i gz8 = {0, 0, 0, 0, 0, 0, 0, 0};
    __builtin_amdgcn_tensor_load_to_lds(g0, g1, gz4, gz4, gz8, 0);
    __builtin_amdgcn_s_wait_tensorcnt(0);
  }
  __syncthreads();

  // The TDM write to LDS is invisible to the compiler (descriptor holds only
  // integers). Escape sB's address into a memory-clobbering asm so the
  // ds_loads below cannot be folded to poison and eliminated.
  asm volatile("" : : "r"(&sB[0]) : "memory");

  if (m0 >= M) return;

  const float* arow = A + (size_t)(m0 + l) * F;   // A row for this lane
  v8f c = {};

#pragma unroll
  for (int k0 = 0; k0 < F; k0 += 4) {
    // A fragment: K = k0 + 2*hf + {0,1} (8B global load, L0-resident reuse)
    v2f a = *(const v2f*)(arow + k0 + 2 * hf);
    // B fragment from LDS (padded stride => conflict-free halves)
    v2f b;
    b.x = sB[(k0 + 2 * hf) * BSTRIDE + n0 + l];
    b.y = sB[(k0 + 2 * hf + 1) * BSTRIDE + n0 + l];
    c = __builtin_amdgcn_wmma_f32_16x16x4_f32(
        /*neg_a=*/false, a, /*neg_b=*/false, b,
        /*c_mod=*/(short)0, c, /*reuse_a=*/false, /*reuse_b=*/false);
  }

#pragma unroll
  for (int r = 0; r < 8; ++r) {
    C[(size_t)(m0 + r + 8 * hf) * F + n0 + l] = c[r];
  }
}

// ---------------------------------------------------------------------------
// Edge-parallel SpMM:  out[row] += val * (filt ? filt[col] : 1) * src[col]
// One wave per edge: 32 lanes x float4 = one 128-float row, coalesced 512B
// gather (L2-resident), scatter via fire-and-forget device-scope
// global_atomic_add_f32 (STOREcnt, no return). Edge metadata forced uniform
// (readfirstlane) -> scalar SMEM loads.
// ---------------------------------------------------------------------------
__global__ void spmm_edges(const int* __restrict__ rows,
                           const int* __restrict__ cols,
                           const float* __restrict__ val,
                           const float* __restrict__ src,
                           const float* __restrict__ filt,
                           float* __restrict__ out, int E) {
  const int lane = threadIdx.x & 31;
  const int gw = blockIdx.x * (blockDim.x >> 5) + (threadIdx.x >> 5);
  const int nw = gridDim.x * (blockDim.x >> 5);

  for (int e0 = gw; e0 < E; e0 += nw) {
    const int e = __builtin_amdgcn_readfirstlane(e0);   // wave-uniform edge id

    // prefetch next iteration's edge metadata (global_prefetch_b8)
    __builtin_prefetch(cols + e + nw, 0, 0);
    __builtin_prefetch(val + e + nw, 0, 0);

    const int   r = rows[e];
    const int   c = cols[e];
    float       v = val[e];
    if (filt) v *= filt[c];                    // fused per-row filter (pass 2)

    const float4 xv = ((const float4*)(src + (size_t)c * F))[lane];
    float* o = out + (size_t)r * F + (size_t)lane * 4;

    asm volatile(
        "global_atomic_add_f32 %0, %1, off scope:SCOPE_DEV\n\t"
        "global_atomic_add_f32 %0, %2, off offset:4 scope:SCOPE_DEV\n\t"
        "global_atomic_add_f32 %0, %3, off offset:8 scope:SCOPE_DEV\n\t"
        "global_atomic_add_f32 %0, %4, off offset:12 scope:SCOPE_DEV"
        :
        : "v"(o), "v"(v * xv.x), "v"(v * xv.y), "v"(v * xv.z), "v"(v * xv.w)
        : "memory");
  }
}

// ---------------------------------------------------------------------------
// Orchestration:
//   out = 0
//   y   = x @ W                      (WMMA + TDM-staged W)
//   for each branch b in {1,2,3}:
//     t = 0
//     t   = SpMM(A_b, y)             (pass 1)
//     out += SpMM(A_b, filt_b * t)   (pass 2, filter fused into gather,
//                                     branch sum fused via atomics)
// ---------------------------------------------------------------------------
extern "C" void kernel_launch(void* const* d_in, const int* in_sizes, int n_in,
                              void* d_out, int out_size, void* d_ws, size_t ws_size,
                              hipStream_t stream) {
  const float* x  = (const float*)d_in[0];
  const float* w  = (const float*)d_in[1];
  const float* filts[3] = {(const float*)d_in[2], (const float*)d_in[3],
                           (const float*)d_in[4]};
  const int*   idxs[3]  = {(const int*)d_in[5], (const int*)d_in[7],
                           (const int*)d_in[9]};
  const float* vals[3]  = {(const float*)d_in[6], (const float*)d_in[8],
                           (const float*)d_in[10]};

  const int M = in_sizes[0] / F;   // 100000 nodes
  const int E = in_sizes[6];       // 3.2M edges per branch

  float* y   = (float*)d_ws;                 // [M,F] transformed features
  float* t   = y + (size_t)M * F;            // [M,F] per-branch intermediate
  float* out = (float*)d_out;
  const long nf = (long)M * F;

  zero_kernel<<<2048, 256, 0, stream>>>(out, nf);
  gemm_wmma_f32<<<(M + 15) / 16, 256, 0, stream>>>(x, w, y, M);

  const int waves_per_block = 256 / 32;
  int blocks = (E + waves_per_block - 1) / waves_per_block;
  if (blocks > 8192) blocks = 8192;

  for (int b = 0; b < 3; ++b) {
    zero_kernel<<<2048, 256, 0, stream>>>(t, nf);
    // pass 1: t = A_b * y
    spmm_edges<<<blocks, 256, 0, stream>>>(idxs[b], idxs[b] + E, vals[b],
                                           y, nullptr, t, E);
    // pass 2: out += A_b * (filt_b (.) t)
    spmm_edges<<<blocks, 256, 0, stream>>>(idxs[b], idxs[b] + E, vals[b],
                                           t, filts[b], out, E);
  }
}